// GCNModel_45569603010897
// MI455X (gfx1250) — compile-verified
//
#include <hip/hip_runtime.h>
#include <hip/hip_bf16.h>

typedef __attribute__((ext_vector_type(2))) float v2f;
typedef __attribute__((ext_vector_type(8))) float v8f;

#define DIN 128
#define DCOL 64   // DH == DOUT == 64

// ---------------- degree / dinv ----------------
__global__ void k_init_deg(float* __restrict__ deg, int n) {
    int i = blockIdx.x * blockDim.x + threadIdx.x;
    if (i < n) deg[i] = 1.0f;            // self-loop contributes 1
}

__global__ void k_accum_deg(const int* __restrict__ dst, float* __restrict__ deg, int e) {
    int i = blockIdx.x * blockDim.x + threadIdx.x;
    if (i < e) atomicAdd(&deg[dst[i]], 1.0f);
}

__global__ void k_finish_dinv(float* __restrict__ deg, int n) {
    int i = blockIdx.x * blockDim.x + threadIdx.x;
    if (i < n) deg[i] = rsqrtf(deg[i]);  // deg >= 1 always
}

// ---------------- fp32 WMMA GEMM: H[n x 64] = (relu?)A[n x K] @ W[K x 64] (+bias) ----
// grid.x = n/16 row tiles, block = 128 threads (4 waves, one 16-col tile per wave).
// K / RELU / BIAS are compile-time so the k-loop unrolls into clean
// load-clause + v_wmma_f32_16x16x4_f32 chains with no per-element selects.
//
// V_WMMA_F32_16X16X4_F32 operand layout (wave32):
//   A (16x4): lanes 0-15 rows M=0..15, vgpr0/1 = K0/K1; lanes 16-31 = K2/K3
//   B (4x16): lanes 0-15 cols N=0..15, vgpr0/1 = K0/K1; lanes 16-31 = K2/K3
//   C/D (16x16): vgpr v, lanes 0-15 -> M=v, N=lane; lanes 16-31 -> M=v+8
template <int K, bool RELU, bool BIAS>
__global__ __launch_bounds__(128)
void k_gemm_wmma(const float* __restrict__ A, const float* __restrict__ W,
                 float* __restrict__ H, const float* __restrict__ bias) {
    const int lane    = threadIdx.x & 31;
    const int wave    = threadIdx.x >> 5;        // 0..3 -> column tile
    const int rowBase = blockIdx.x << 4;
    const int colBase = wave << 4;
    const int half    = lane >> 4;               // 0: K=[0,1], 1: K=[2,3]
    const int l16     = lane & 15;

    const float* __restrict__ arow = A + (size_t)(rowBase + l16) * K;
    const float* __restrict__ wcol = W + colBase + l16;

    v8f c = {};
    #pragma unroll 8
    for (int kb = 0; kb < K; kb += 4) {
        const int k0 = kb + half * 2;
        float a0 = arow[k0];
        float a1 = arow[k0 + 1];
        if constexpr (RELU) { a0 = fmaxf(a0, 0.0f); a1 = fmaxf(a1, 0.0f); }
        v2f a; a[0] = a0; a[1] = a1;
        v2f b; b[0] = wcol[(size_t)k0 * DCOL]; b[1] = wcol[(size_t)(k0 + 1) * DCOL];
        c = __builtin_amdgcn_wmma_f32_16x16x4_f32(
                /*neg_a=*/false, a, /*neg_b=*/false, b,
                /*c_mod=*/(short)0, c, /*reuse_a=*/false, /*reuse_b=*/false);
    }

    float bv;
    if constexpr (BIAS) bv = bias[colBase + l16];
    float* __restrict__ hout = H + colBase + l16;
    #pragma unroll
    for (int v = 0; v < 8; ++v) {
        const int row = rowBase + v + half * 8;
        float val = c[v];
        if constexpr (BIAS) val += bv;      // no dead "+0" adds in non-bias variants
        hout[(size_t)row * DCOL] = val;
    }
}

// ---------------- aggregation: self-loop init (no atomics) ----------------
// out[i,:] = dinv[i]^2 * h[i,:] + bias   (float4-vectorized, 16 threads/node)
__global__ void k_agg_init(const float* __restrict__ h, const float* __restrict__ dinv,
                           const float* __restrict__ bias, float* __restrict__ out, int n) {
    int t = blockIdx.x * blockDim.x + threadIdx.x;
    int i = t >> 4, q = t & 15;
    if (i >= n) return;
    float w = dinv[i]; w *= w;
    float4 hv = reinterpret_cast<const float4*>(h + (size_t)i * DCOL)[q];
    float4 bv = reinterpret_cast<const float4*>(bias)[q];
    float4 o;
    o.x = fmaf(w, hv.x, bv.x); o.y = fmaf(w, hv.y, bv.y);
    o.z = fmaf(w, hv.z, bv.z); o.w = fmaf(w, hv.w, bv.w);
    reinterpret_cast<float4*>(out + (size_t)i * DCOL)[q] = o;
}

// ---------------- aggregation: edge scatter-add (L2-resident atomics) ----------------
// 16 threads per edge, each gathers float4 of h[src] and does 4 f32 atomics to out[dst].
__global__ void k_agg_edges(const int* __restrict__ src, const int* __restrict__ dst,
                            const float* __restrict__ dinv, const float* __restrict__ h,
                            float* __restrict__ out, int e) {
    int t = blockIdx.x * blockDim.x + threadIdx.x;
    int ed = t >> 4, q = t & 15;
    if (ed >= e) return;
    int s = src[ed];
    int d = dst[ed];
    float w = dinv[s] * dinv[d];
    float4 hv = reinterpret_cast<const float4*>(h + (size_t)s * DCOL)[q];
    float* o = out + (size_t)d * DCOL + q * 4;
    atomicAdd(o + 0, w * hv.x);
    atomicAdd(o + 1, w * hv.y);
    atomicAdd(o + 2, w * hv.z);
    atomicAdd(o + 3, w * hv.w);
}

extern "C" void kernel_launch(void* const* d_in, const int* in_sizes, int n_in,
                              void* d_out, int out_size, void* d_ws, size_t ws_size,
                              hipStream_t stream) {
    const float* x   = (const float*)d_in[0];
    const int*   ei  = (const int*)  d_in[1];
    const float* W1  = (const float*)d_in[2];
    const float* b1  = (const float*)d_in[3];
    const float* W2  = (const float*)d_in[4];
    const float* b2  = (const float*)d_in[5];
    const float* Wfc = (const float*)d_in[6];
    const float* bfc = (const float*)d_in[7];
    float* out = (float*)d_out;

    const int N = in_sizes[0] / DIN;      // 50000
    const int E = in_sizes[1] / 2;        // 800000
    const int* src = ei;
    const int* dst = ei + E;

    // workspace carve-up (26 MB): dinv | buf0 (h) | buf1 (agg)
    float* dinv = (float*)d_ws;
    float* buf0 = dinv + ((N + 255) / 256) * 256;
    float* buf1 = buf0 + (size_t)N * DCOL;

    const int blk = 256;
    const int gN   = (N + blk - 1) / blk;
    const int gE   = (E + blk - 1) / blk;
    const int gN16 = (N * 16 + blk - 1) / blk;   // 16 threads per node
    const int gE16 = (E * 16 + blk - 1) / blk;   // 16 threads per edge
    const int gGemm = N / 16;                    // 50000 % 16 == 0

    // degrees -> dinv (in place)
    k_init_deg   <<<gN, blk, 0, stream>>>(dinv, N);
    k_accum_deg  <<<gE, blk, 0, stream>>>(dst, dinv, E);
    k_finish_dinv<<<gN, blk, 0, stream>>>(dinv, N);

    // layer 1: h1 = x @ W1 ; agg1 = scatter(norm * h1) + b1   (ReLU deferred)
    k_gemm_wmma<DIN,  false, false><<<gGemm, 128, 0, stream>>>(x,    W1,  buf0, nullptr);
    k_agg_init  <<<gN16, blk, 0, stream>>>(buf0, dinv, b1, buf1, N);
    k_agg_edges <<<gE16, blk, 0, stream>>>(src, dst, dinv, buf0, buf1, E);

    // layer 2: h2 = relu(agg1) @ W2 ; agg2 = scatter(norm * h2) + b2
    k_gemm_wmma<DCOL, true,  false><<<gGemm, 128, 0, stream>>>(buf1, W2,  buf0, nullptr);
    k_agg_init  <<<gN16, blk, 0, stream>>>(buf0, dinv, b2, buf1, N);
    k_agg_edges <<<gE16, blk, 0, stream>>>(src, dst, dinv, buf0, buf1, E);

    // head: out = agg2 @ Wfc + bfc
    k_gemm_wmma<DCOL, false, true ><<<gGemm, 128, 0, stream>>>(buf1, Wfc, out,  bfc);
}